// GAT_67869073212170
// MI455X (gfx1250) — compile-verified
//
#include <hip/hip_runtime.h>

// ---------------------------------------------------------------------------
// Problem constants (match reference)
// ---------------------------------------------------------------------------
#define NG   512                 // graphs
#define NPG  152                 // nodes per graph
#define DEG  8
#define NN   (NG * NPG)          // 77824 nodes
#define EE   (NN * DEG)          // 622592 edges (before self loops)
#define FIN  128
#define HEADS 6
#define DH   32
#define CD   192                 // HEADS * DH
#define EPG  (NPG * DEG)         // 1216 edges per graph
#define TOTE (EPG + NPG)         // + self loops
#define NEG_SLOPE 0.2f
#define GN_EPS 1e-5f

typedef __attribute__((ext_vector_type(2))) float v2f;
typedef __attribute__((ext_vector_type(8))) float v8f;

// ---------------------------------------------------------------------------
// GEMM: Hout[N, 192] = X[N, K] @ W[K, 192]   (fp32 WMMA 16x16x4)
// Block = 128 threads = 4 wave32s. Each block: one 64-row x 16-col tile.
// W column slab (K x 16) staged in LDS, shared by all 4 waves.
// ---------------------------------------------------------------------------
template <int K>
__global__ __launch_bounds__(128) void gat_gemm(const float* __restrict__ X,
                                                const float* __restrict__ W,
                                                float* __restrict__ Hout) {
    __shared__ float Wl[CD * 16];          // max K = 192 -> 12 KB
    const int t = threadIdx.x;
    const int nbase = blockIdx.y * 16;

    for (int idx = t; idx < K * 16; idx += 128) {
        const int k = idx >> 4, n = idx & 15;
        Wl[idx] = W[k * CD + nbase + n];
    }
    __syncthreads();

    const int lane = t & 31;               // wave32
    const int wave = t >> 5;
    const int mbase = (blockIdx.x * 4 + wave) * 16;
    const int klo = (lane >> 4) * 2;       // lanes 16-31 hold K+2/K+3 halves
    const int nlo = lane & 15;

    const float* xrow = X + (size_t)(mbase + nlo) * K;

    v8f acc = {0.f, 0.f, 0.f, 0.f, 0.f, 0.f, 0.f, 0.f};
    #pragma unroll 4
    for (int kk = 0; kk < K; kk += 4) {
        v2f a, b;
        // A 16x4 f32: lane L<16 -> row L, {K=kk, kk+1}; lane>=16 -> {kk+2, kk+3}
        a.x = xrow[kk + klo];
        a.y = xrow[kk + klo + 1];
        // B 4x16 f32: row striped across lanes within VGPR, mirrored halves
        b.x = Wl[(kk + klo) * 16 + nlo];
        b.y = Wl[(kk + klo + 1) * 16 + nlo];
        acc = __builtin_amdgcn_wmma_f32_16x16x4_f32(
            false, a, false, b, (short)0, acc, false, false);
    }

    // C/D layout: VGPR i -> M = i + (lane>=16 ? 8 : 0), N = lane & 15
    const int mo = (lane >> 4) * 8;
    float* orow = Hout + (size_t)(mbase + mo) * CD + nbase + nlo;
    #pragma unroll
    for (int i = 0; i < 8; i++) orow[(size_t)i * CD] = acc[i];
}

// ---------------------------------------------------------------------------
// Monotone float <-> uint encoding for atomic segment-max
// ---------------------------------------------------------------------------
__device__ __forceinline__ unsigned fenc(float x) {
    unsigned u = __float_as_uint(x);
    return (u & 0x80000000u) ? ~u : (u | 0x80000000u);
}
__device__ __forceinline__ float fdec(unsigned u) {
    return (u & 0x80000000u) ? __uint_as_float(u & 0x7FFFFFFFu)
                             : __uint_as_float(~u);
}

__device__ __forceinline__ void edge_logit(const int* __restrict__ esrc,
                                           const int* __restrict__ edst,
                                           int g, int base, int e, int hh,
                                           const float* __restrict__ als,
                                           const float* __restrict__ ald,
                                           int& sl, int& dl, float& v) {
    if (e < EPG) {
        sl = esrc[(size_t)g * EPG + e] - base;
        dl = edst[(size_t)g * EPG + e] - base;
    } else {
        sl = dl = e - EPG;                 // self loop
    }
    v = als[sl * HEADS + hh] + ald[dl * HEADS + hh];
    v = v > 0.f ? v : NEG_SLOPE * v;       // leaky relu
}

// ---------------------------------------------------------------------------
// Fused per-graph kernel: attention logits -> segment softmax -> aggregation
// -> bias -> ELU -> GraphNorm. One workgroup per graph; everything in LDS
// (~248 KB of the WGP's 320 KB), all atomics are LDS (ds_*) atomics.
// ---------------------------------------------------------------------------
__global__ __launch_bounds__(256) void gat_graph(
    const float* __restrict__ Hin, const int* __restrict__ esrc,
    const int* __restrict__ edst, const float* __restrict__ a_src,
    const float* __restrict__ a_dst, const float* __restrict__ bias,
    const float* __restrict__ gw, const float* __restrict__ gb,
    const float* __restrict__ gms, float* __restrict__ Out) {
    extern __shared__ float smem[];
    float* h_l     = smem;                       // NPG*CD  (114 KB)
    float* acc     = h_l + NPG * CD;             // NPG*CD  (114 KB)
    float* als     = acc + NPG * CD;             // NPG*HEADS
    float* ald     = als + NPG * HEADS;          // NPG*HEADS
    float* denom   = ald + NPG * HEADS;          // NPG*HEADS
    unsigned* mkey = (unsigned*)(denom + NPG * HEADS); // NPG*HEADS

    const int g = blockIdx.x;
    const int t = threadIdx.x;
    const int base = g * NPG;
    const size_t gbase = (size_t)base * CD;

    // Stage this graph's features; zero accumulators
    for (int i = t; i < NPG * CD; i += 256) {
        h_l[i] = Hin[gbase + i];
        acc[i] = 0.f;
    }
    __syncthreads();

    // Per-(node, head) attention projections
    for (int i = t; i < NPG * HEADS; i += 256) {
        const int n = i / HEADS, hh = i % HEADS;
        const float* hr = &h_l[n * CD + hh * DH];
        float s = 0.f, d = 0.f;
        #pragma unroll
        for (int j = 0; j < DH; j++) {
            s += hr[j] * a_src[hh * DH + j];
            d += hr[j] * a_dst[hh * DH + j];
        }
        als[i] = s;
        ald[i] = d;
        denom[i] = 0.f;
        mkey[i] = 0u;                      // below fenc of any finite float
    }
    __syncthreads();

    // Pass 1: segment max over incoming edges of each dst
    for (int i = t; i < TOTE * HEADS; i += 256) {
        int sl, dl; float v;
        edge_logit(esrc, edst, g, base, i / HEADS, i % HEADS, als, ald, sl, dl, v);
        atomicMax(&mkey[dl * HEADS + (i % HEADS)], fenc(v));
    }
    __syncthreads();

    // Pass 2: softmax denominators
    for (int i = t; i < TOTE * HEADS; i += 256) {
        int sl, dl; float v;
        const int hh = i % HEADS;
        edge_logit(esrc, edst, g, base, i / HEADS, hh, als, ald, sl, dl, v);
        const float m = fdec(mkey[dl * HEADS + hh]);
        atomicAdd(&denom[dl * HEADS + hh], __expf(v - m));
    }
    __syncthreads();

    // Pass 3: weighted aggregation acc[dst] += alpha * h[src]
    for (int i = t; i < TOTE * HEADS; i += 256) {
        int sl, dl; float v;
        const int hh = i % HEADS;
        edge_logit(esrc, edst, g, base, i / HEADS, hh, als, ald, sl, dl, v);
        const float m = fdec(mkey[dl * HEADS + hh]);
        const float alpha = __expf(v - m) / denom[dl * HEADS + hh];
        const float* hs = &h_l[sl * CD + hh * DH];
        float* ad = &acc[dl * CD + hh * DH];
        #pragma unroll
        for (int j = 0; j < DH; j++) atomicAdd(&ad[j], alpha * hs[j]);
    }
    __syncthreads();

    // Bias + ELU (in place)
    for (int i = t; i < NPG * CD; i += 256) {
        const float v = acc[i] + bias[i % CD];
        acc[i] = v > 0.f ? v : (__expf(v) - 1.f);
    }
    __syncthreads();

    // GraphNorm stats: per (graph, channel) over 152 nodes. Reuse als/ald.
    if (t < CD) {
        float s = 0.f;
        for (int n = 0; n < NPG; n++) s += acc[n * CD + t];
        const float mm = (s / (float)NPG) * gms[t];
        float vs = 0.f;
        for (int n = 0; n < NPG; n++) {
            const float o = acc[n * CD + t] - mm;
            vs += o * o;
        }
        const float rstd = rsqrtf(vs / (float)NPG + GN_EPS);
        als[t] = mm;                       // shifted mean
        ald[t] = rstd * gw[t];             // fused scale
    }
    __syncthreads();

    for (int i = t; i < NPG * CD; i += 256) {
        const int c = i % CD;
        Out[gbase + i] = (acc[i] - als[c]) * ald[c] + gb[c];
    }
}

// ---------------------------------------------------------------------------
// Launch: gemm1 -> graph1 -> gemm2 -> graph2
// ---------------------------------------------------------------------------
extern "C" void kernel_launch(void* const* d_in, const int* in_sizes, int n_in,
                              void* d_out, int out_size, void* d_ws, size_t ws_size,
                              hipStream_t stream) {
    const float* x    = (const float*)d_in[0];
    const int*   ei   = (const int*)d_in[1];
    // d_in[2] = batch (implicit in per-graph blocking)
    const float* W1   = (const float*)d_in[3];
    const float* as1  = (const float*)d_in[4];
    const float* ad1  = (const float*)d_in[5];
    const float* b1   = (const float*)d_in[6];
    const float* W2   = (const float*)d_in[7];
    const float* as2  = (const float*)d_in[8];
    const float* ad2  = (const float*)d_in[9];
    const float* b2   = (const float*)d_in[10];
    const float* gw1  = (const float*)d_in[11];
    const float* gb1  = (const float*)d_in[12];
    const float* gms1 = (const float*)d_in[13];
    const float* gw2  = (const float*)d_in[14];
    const float* gb2  = (const float*)d_in[15];
    const float* gms2 = (const float*)d_in[16];

    const int* esrc = ei;          // edge_index[0]
    const int* edst = ei + EE;     // edge_index[1]

    float* bufA = (float*)d_ws;                    // [N, 192]
    float* bufB = bufA + (size_t)NN * CD;          // [N, 192]
    float* outF = (float*)d_out;

    const size_t smem = (size_t)(2 * NPG * CD + 4 * NPG * HEADS) * sizeof(float);
    const dim3 ggrid(NN / 64, CD / 16);            // (1216, 12)

    gat_gemm<FIN><<<ggrid, 128, 0, stream>>>(x, W1, bufA);
    gat_graph<<<NG, 256, smem, stream>>>(bufA, esrc, edst, as1, ad1, b1,
                                         gw1, gb1, gms1, bufB);
    gat_gemm<CD><<<ggrid, 128, 0, stream>>>(bufB, W2, bufA);
    gat_graph<<<NG, 256, smem, stream>>>(bufA, esrc, edst, as2, ad2, b2,
                                         gw2, gb2, gms2, outF);
}